// GCNLayer_54863912239191
// MI455X (gfx1250) — compile-verified
//
#include <hip/hip_runtime.h>

typedef float v2f __attribute__((ext_vector_type(2)));
typedef float v8f __attribute__((ext_vector_type(8)));

#define FDIM 32

// ---------------- helpers ----------------

__global__ void zero_f32_kernel(float* __restrict__ p, int n) {
    int i = blockIdx.x * blockDim.x + threadIdx.x;
    if (i < n) p[i] = 0.0f;
}

// deg[src] += 1 per edge
__global__ void count_deg_kernel(const int* __restrict__ srcs,
                                 float* __restrict__ deg, int E) {
    int e = blockIdx.x * blockDim.x + threadIdx.x;
    if (e < E) unsafeAtomicAdd(&deg[srcs[e]], 1.0f);
}

// deg -> clip(deg,1)^-0.5 in place
__global__ void deg_to_norm_kernel(float* __restrict__ p, int n) {
    int i = blockIdx.x * blockDim.x + threadIdx.x;
    if (i < n) p[i] = rsqrtf(fmaxf(p[i], 1.0f));
}

// out[n*32+f] = bias[f]
__global__ void init_out_kernel(const float* __restrict__ bias,
                                float* __restrict__ out, int total) {
    int i = blockIdx.x * blockDim.x + threadIdx.x;
    if (i < total) out[i] = bias[i & (FDIM - 1)];
}

// ---------------- dense transform: H = F(ROWSx32) @ K(32x32) via f32 WMMA ----------------
// One wave computes a 16-row x 32-col tile with 16x V_WMMA_F32_16X16X4_F32.
// Block = 64 threads = 2 waves = 32 rows. Grid covers N32 = (N/32)*32 rows exactly,
// so EXEC is all-ones for every wave that issues WMMA (ISA requirement).
__global__ void __launch_bounds__(64)
gemm_wmma_kernel(const float* __restrict__ F, const float* __restrict__ K,
                 float* __restrict__ H) {
    const int lane = threadIdx.x & 31;
    const int wave = threadIdx.x >> 5;
    const int row_base = (blockIdx.x * 2 + wave) * 16;
    const int half = lane >> 4;   // 0: lanes 0-15, 1: lanes 16-31
    const int l16  = lane & 15;

    v8f c0 = {};   // cols 0..15
    v8f c1 = {};   // cols 16..31

    const float* __restrict__ arow = F + (size_t)(row_base + l16) * FDIM;

#pragma unroll
    for (int kb = 0; kb < 8; ++kb) {
        // A 16x4 f32 layout: lanes 0-15 -> K = kb*4 + {0,1}; lanes 16-31 -> K = kb*4 + {2,3}
        const int k0 = kb * 4 + half * 2;
        v2f a;
        a.x = arow[k0];
        a.y = arow[k0 + 1];

        // B 4x16 f32 (transposed striping): v[0] holds K=k0, v[1] holds K=k0+1, N = l16
        v2f b0, b1;
        b0.x = K[k0 * FDIM + l16];
        b0.y = K[(k0 + 1) * FDIM + l16];
        b1.x = K[k0 * FDIM + 16 + l16];
        b1.y = K[(k0 + 1) * FDIM + 16 + l16];

        c0 = __builtin_amdgcn_wmma_f32_16x16x4_f32(false, a, false, b0,
                                                   (short)0, c0, false, false);
        c1 = __builtin_amdgcn_wmma_f32_16x16x4_f32(false, a, false, b1,
                                                   (short)0, c1, false, false);
    }

    // C/D 16x16 f32 layout: VGPR v -> M = v + half*8, N = l16
    float* __restrict__ hr = H + (size_t)row_base * FDIM;
#pragma unroll
    for (int v = 0; v < 8; ++v) {
        const int m = v + half * 8;
        hr[m * FDIM + l16]      = c0[v];
        hr[m * FDIM + 16 + l16] = c1[v];
    }
}

// scalar tail for rows not covered by the WMMA kernel (empty for N=100000)
__global__ void gemm_tail_kernel(const float* __restrict__ F,
                                 const float* __restrict__ K,
                                 float* __restrict__ H, int row0, int nrows) {
    int t = blockIdx.x * blockDim.x + threadIdx.x;
    int r = row0 + (t >> 5);
    int c = t & 31;
    if (r >= row0 + nrows) return;
    float acc = 0.0f;
    const float* fr = F + (size_t)r * FDIM;
#pragma unroll
    for (int k = 0; k < FDIM; ++k) acc += fr[k] * K[k * FDIM + c];
    H[(size_t)r * FDIM + c] = acc;
}

// ---------------- edge scatter ----------------
// 8 lanes per edge; each lane: one 16B load from h[dst], 4 f32 atomics into out[src].
__global__ void __launch_bounds__(256)
scatter_kernel(const int* __restrict__ srcs, const int* __restrict__ dsts,
               const float* __restrict__ wts, const float* __restrict__ norm,
               const float* __restrict__ h, float* __restrict__ out, int E) {
    int t = blockIdx.x * blockDim.x + threadIdx.x;
    int e = t >> 3;
    int c = t & 7;
    if (e >= E) return;
    const int s = srcs[e];
    const int d = dsts[e];
    const float w = norm[s] * wts[e] * norm[d];
    const float4 hv = *reinterpret_cast<const float4*>(h + (size_t)d * FDIM + c * 4);
    float* o = out + (size_t)s * FDIM + c * 4;
    unsafeAtomicAdd(o + 0, hv.x * w);
    unsafeAtomicAdd(o + 1, hv.y * w);
    unsafeAtomicAdd(o + 2, hv.z * w);
    unsafeAtomicAdd(o + 3, hv.w * w);
}

// ---------------- launch ----------------

extern "C" void kernel_launch(void* const* d_in, const int* in_sizes, int n_in,
                              void* d_out, int out_size, void* d_ws, size_t ws_size,
                              hipStream_t stream) {
    const float* features = (const float*)d_in[0];
    const int*   srcs     = (const int*)d_in[1];
    const int*   dsts     = (const int*)d_in[2];
    const float* wts      = (const float*)d_in[3];
    const float* kern     = (const float*)d_in[4];
    const float* bias     = (const float*)d_in[5];
    float* out = (float*)d_out;

    const int N = in_sizes[0] / FDIM;
    const int E = in_sizes[1];

    // workspace: norm[N] then h[N*32]
    float* deg = (float*)d_ws;
    float* h   = deg + N;

    // 1) degrees
    zero_f32_kernel<<<(N + 255) / 256, 256, 0, stream>>>(deg, N);
    count_deg_kernel<<<(E + 255) / 256, 256, 0, stream>>>(srcs, deg, E);
    deg_to_norm_kernel<<<(N + 255) / 256, 256, 0, stream>>>(deg, N);

    // 2) dense transform via WMMA (32 rows per 64-thread block)
    const int N32 = (N / 32) * 32;
    if (N32 > 0)
        gemm_wmma_kernel<<<N32 / 32, 64, 0, stream>>>(features, kern, h);
    if (N - N32 > 0) {
        const int tail = N - N32;
        gemm_tail_kernel<<<(tail * FDIM + 255) / 256, 256, 0, stream>>>(
            features, kern, h, N32, tail);
    }

    // 3) out = bias (broadcast), then scatter-accumulate edges
    init_out_kernel<<<(N * FDIM + 255) / 256, 256, 0, stream>>>(bias, out, N * FDIM);

    const long long tot = (long long)E * 8;
    scatter_kernel<<<(unsigned)((tot + 255) / 256), 256, 0, stream>>>(
        srcs, dsts, wts, deg /* now norm */, h, out, E);
}